// MultiBoxLoss_68315749810846
// MI455X (gfx1250) — compile-verified
//
#include <hip/hip_runtime.h>
#include <math.h>
#include <stdint.h>

#define WAVE 32
#define NEG_INF (-__builtin_inff())

#if defined(__has_builtin)
#if __has_builtin(__builtin_amdgcn_global_load_async_to_lds_b32) && \
    __has_builtin(__builtin_amdgcn_s_wait_asynccnt)
#define HAVE_ASYNC_LDS 1
#endif
#endif

#if defined(HAVE_ASYNC_LDS)
#pragma message("CDNA5: using global_load_async_to_lds + s_wait_asynccnt path")
#else
#pragma message("CDNA5: async-LDS builtins not found; plain LDS staging fallback")
#endif

// Stage n floats (n <= 256) of global data into LDS, visible to whole block.
__device__ __forceinline__ void load_floats_to_lds(const float* __restrict__ g,
                                                   float* s, int n) {
#if defined(HAVE_ASYNC_LDS)
  if ((int)threadIdx.x < WAVE) {
    for (int i = threadIdx.x; i < n; i += WAVE) {
      __builtin_amdgcn_global_load_async_to_lds_b32(
          (__attribute__((address_space(1))) int*)(uintptr_t)(g + i),
          (__attribute__((address_space(3))) int*)(s + i),
          /*offset=*/0, /*cpol=*/0);
    }
    __builtin_amdgcn_s_wait_asynccnt(0);
  }
  __syncthreads();
#else
  if ((int)threadIdx.x < n) s[threadIdx.x] = g[threadIdx.x];
  __syncthreads();
#endif
}

// Fixed-order block (256 threads) reductions. Result valid on tid 0.
__device__ __forceinline__ float blockReduceSumF(float v, float* sm8) {
  __syncthreads();
  for (int o = 16; o > 0; o >>= 1) v += __shfl_xor(v, o, 32);
  int w = threadIdx.x >> 5;
  if ((threadIdx.x & 31) == 0) sm8[w] = v;
  __syncthreads();
  float r = 0.f;
  if (threadIdx.x < 8) r = sm8[threadIdx.x];
  if (threadIdx.x < 32)
    for (int o = 4; o > 0; o >>= 1) r += __shfl_xor(r, o, 32);
  return r;
}
__device__ __forceinline__ int blockReduceSumI(int v, int* sm8) {
  __syncthreads();
  for (int o = 16; o > 0; o >>= 1) v += __shfl_xor(v, o, 32);
  int w = threadIdx.x >> 5;
  if ((threadIdx.x & 31) == 0) sm8[w] = v;
  __syncthreads();
  int r = 0;
  if (threadIdx.x < 8) r = sm8[threadIdx.x];
  if (threadIdx.x < 32)
    for (int o = 4; o > 0; o >>= 1) r += __shfl_xor(r, o, 32);
  return r;
}

__global__ void k_init(unsigned long long* bpk, int* num_pos, int nk, int B) {
  int i = blockIdx.x * blockDim.x + threadIdx.x;
  if (i < nk) bpk[i] = 0ull;
  if (i < B) num_pos[i] = 0;
}

// Kernel 1: per (b,p) best-truth overlap/idx; per (b,j) best prior via packed
// u64 atomicMax (key = overlap_bits<<32 | (~p) -> max == first-occurrence argmax).
__global__ void __launch_bounds__(256) k_match(
    const float* __restrict__ priors, const float* __restrict__ targets,
    float* __restrict__ bto, int* __restrict__ bti,
    unsigned long long* __restrict__ bpk, int P, int NOBJ) {
  __shared__ float tr[256];
  int b = blockIdx.y;
  load_floats_to_lds(targets + (size_t)b * NOBJ * 5, tr, NOBJ * 5);
  int p = blockIdx.x * 256 + threadIdx.x;
  bool valid = p < P;
  float cx = 0.f, cy = 0.f, w = 1.f, h = 1.f;
  if (valid) {
    const float* pr = priors + (size_t)p * 4;
    cx = pr[0]; cy = pr[1]; w = pr[2]; h = pr[3];
  }
  float ax1 = cx - w * 0.5f, ay1 = cy - h * 0.5f;
  float ax2 = cx + w * 0.5f, ay2 = cy + h * 0.5f;
  float area_p = (ax2 - ax1) * (ay2 - ay1);
  float best = -1.f;
  int bj = 0;
  for (int j = 0; j < NOBJ; ++j) {
    float tx1 = tr[j * 5 + 0], ty1 = tr[j * 5 + 1];
    float tx2 = tr[j * 5 + 2], ty2 = tr[j * 5 + 3];
    float iw = fmaxf(fminf(ax2, tx2) - fmaxf(ax1, tx1), 0.f);
    float ih = fmaxf(fminf(ay2, ty2) - fmaxf(ay1, ty1), 0.f);
    float inter = iw * ih;
    float at = (tx2 - tx1) * (ty2 - ty1);
    float ov = inter / (at + area_p - inter + 1e-10f);
    if (valid && ov > best) { best = ov; bj = j; }  // strict > : first max
    unsigned long long key = 0ull;
    if (valid)
      key = ((unsigned long long)__float_as_uint(ov) << 32) |
            (unsigned long long)(0xFFFFFFFFu - (unsigned)p);
    for (int o = 16; o > 0; o >>= 1) {
      unsigned long long other = __shfl_xor(key, o, 32);
      if (other > key) key = other;
    }
    if ((threadIdx.x & 31) == 0) atomicMax(&bpk[b * NOBJ + j], key);
  }
  if (valid) {
    size_t i = (size_t)b * P + p;
    bto[i] = best;
    bti[i] = bj;
  }
}

// Kernel 2: forced matches (in-order per batch -> last j wins on duplicates).
__global__ void k_force(const unsigned long long* __restrict__ bpk,
                        float* __restrict__ bto, int* __restrict__ bti,
                        int P, int NOBJ, int B) {
  int b = blockIdx.x * blockDim.x + threadIdx.x;
  if (b >= B) return;
  for (int j = 0; j < NOBJ; ++j) {
    unsigned long long key = bpk[b * NOBJ + j];
    unsigned p = 0xFFFFFFFFu - (unsigned)(key & 0xFFFFFFFFull);
    size_t i = (size_t)b * P + (size_t)p;
    bto[i] = 2.0f;
    bti[i] = j;
  }
}

// Kernel 3: conf_t (overwrites bti in place), encode(), smooth-L1 partials,
// integer num_pos atomics.
__global__ void __launch_bounds__(256) k_encode(
    const float* __restrict__ priors, const float* __restrict__ targets,
    const float* __restrict__ loc_data, const float* __restrict__ bto,
    int* __restrict__ bti_conf, float* __restrict__ partial_l,
    int* __restrict__ num_pos, int P, int NOBJ) {
  __shared__ float tr[256];
  __shared__ float smf[8];
  __shared__ int smi[8];
  int b = blockIdx.y;
  load_floats_to_lds(targets + (size_t)b * NOBJ * 5, tr, NOBJ * 5);
  int p = blockIdx.x * 256 + threadIdx.x;
  float sl = 0.f;
  int posc = 0;
  if (p < P) {
    size_t i = (size_t)b * P + p;
    float ov = bto[i];
    int j = bti_conf[i];
    int cls = 0;
    if (!(ov < 0.5f)) cls = (int)(tr[j * 5 + 4] + 1.0f);
    bti_conf[i] = cls;
    if (cls > 0) {
      posc = 1;
      const float* pr = priors + (size_t)p * 4;
      float cx = pr[0], cy = pr[1], w = pr[2], h = pr[3];
      float x1 = tr[j * 5 + 0], y1 = tr[j * 5 + 1];
      float x2 = tr[j * 5 + 2], y2 = tr[j * 5 + 3];
      float g0 = ((x1 + x2) * 0.5f - cx) / (0.1f * w);
      float g1 = ((y1 + y2) * 0.5f - cy) / (0.1f * h);
      float g2 = logf((x2 - x1) / w + 1e-10f) / 0.2f;
      float g3 = logf((y2 - y1) / h + 1e-10f) / 0.2f;
      float gg[4] = {g0, g1, g2, g3};
      const float* ld = loc_data + i * 4;
#pragma unroll
      for (int c = 0; c < 4; ++c) {
        float d = fabsf(ld[c] - gg[c]);
        sl += (d < 1.f) ? (0.5f * d * d) : (d - 0.5f);
      }
    }
  }
  float slsum = blockReduceSumF(sl, smf);
  int pc = blockReduceSumI(posc, smi);
  if (threadIdx.x == 0) {
    partial_l[(size_t)b * gridDim.x + blockIdx.x] = slsum;
    atomicAdd(&num_pos[b], pc);
  }
}

// Kernel 4: one wave per row. lse = m + log(sum exp(x-m)); gathered via shfl.
// Writes ce_mine, accumulates positive-CE partials. Streams ~509MB.
__global__ void __launch_bounds__(256) k_ce(
    const float* __restrict__ conf, const int* __restrict__ conf_t,
    float* __restrict__ cem, float* __restrict__ partial_c,
    long long R, int C) {
  __shared__ float sm8[8];
  int lane = threadIdx.x & 31;
  long long wid = (((long long)blockIdx.x * blockDim.x) + threadIdx.x) >> 5;
  long long W = ((long long)gridDim.x * blockDim.x) >> 5;
  float acc = 0.f;
  for (long long r = wid; r < R; r += W) {
    const float* base = conf + (size_t)r * C;
    float x0 = (lane < C) ? base[lane] : NEG_INF;
    float x1 = (lane + 32 < C) ? base[lane + 32] : NEG_INF;
    float x2 = (lane + 64 < C) ? base[lane + 64] : NEG_INF;
    if (r + W < R) {  // prefetch this wave's next row (global_prefetch_b8)
      const float* nb = conf + (size_t)(r + W) * C;
      __builtin_prefetch(nb + lane, 0, 1);
      __builtin_prefetch(nb + lane + 32, 0, 1);
      __builtin_prefetch(nb + lane + 64, 0, 1);
    }
    float m = fmaxf(x0, fmaxf(x1, x2));
    for (int o = 16; o > 0; o >>= 1) m = fmaxf(m, __shfl_xor(m, o, 32));
    float s = expf(x0 - m) + expf(x1 - m) + expf(x2 - m);
    for (int o = 16; o > 0; o >>= 1) s += __shfl_xor(s, o, 32);
    float lse = m + logf(s);
    int cls = conf_t[r];
    int chunk = cls >> 5;
    float v = (chunk == 0) ? x0 : ((chunk == 1) ? x1 : x2);
    float g = __shfl(v, cls & 31, 32);
    float ce = lse - g;
    if (lane == 0) {
      bool pos = cls > 0;
      cem[r] = pos ? 0.f : ce;
      if (pos) acc += ce;
    }
  }
  float tot = blockReduceSumF(acc, sm8);
  if (threadIdx.x == 0) partial_c[blockIdx.x] = tot;
}

// Kernel 5: per-batch radix-select (4x 8-bit passes on float bits; all values
// >= 0 so uint order == float order). top-K sum = sum_{>t} + (K-count_gt)*t.
__global__ void __launch_bounds__(256) k_select(
    const float* __restrict__ cem, const int* __restrict__ num_pos,
    float* __restrict__ c_top, int P) {
  __shared__ unsigned hist[256];
  __shared__ unsigned sh_prefix, sh_mask;
  __shared__ int sh_K;
  __shared__ float sf[256];
  __shared__ unsigned sc[256];
  int b = blockIdx.x;
  int tid = threadIdx.x;
  const float* v = cem + (size_t)b * P;
  int np = num_pos[b];
  int K = 3 * np;
  if (K > P - 1) K = P - 1;
  if (K <= 0) {
    if (tid == 0) c_top[b] = 0.f;
    return;
  }
  if (tid == 0) { sh_prefix = 0u; sh_mask = 0u; sh_K = K; }
  __syncthreads();
  for (int pass = 0; pass < 4; ++pass) {
    int shift = 24 - 8 * pass;
    hist[tid] = 0u;
    __syncthreads();
    unsigned mask = sh_mask, prefix = sh_prefix;
    for (int i = tid; i < P; i += 256) {
      unsigned u = __float_as_uint(v[i]);
      if ((u & mask) == prefix) atomicAdd(&hist[(u >> shift) & 255u], 1u);
    }
    __syncthreads();
    if (tid == 0) {
      int Kr = sh_K;
      unsigned cum = 0;
      int sel = 0;
      for (int bin = 255; bin >= 0; --bin) {
        unsigned c = hist[bin];
        if (cum + c >= (unsigned)Kr) { sel = bin; sh_K = Kr - (int)cum; break; }
        cum += c;
      }
      sh_prefix = prefix | ((unsigned)sel << shift);
      sh_mask = mask | (0xFFu << shift);
    }
    __syncthreads();
  }
  unsigned t = sh_prefix;  // exact bits of K-th largest
  float s = 0.f;
  unsigned cnt = 0;
  for (int i = tid; i < P; i += 256) {
    unsigned u = __float_as_uint(v[i]);
    if (u > t) { s += __uint_as_float(u); cnt++; }
  }
  sf[tid] = s; sc[tid] = cnt;
  __syncthreads();
  for (int o = 128; o > 0; o >>= 1) {
    if (tid < o) { sf[tid] += sf[tid + o]; sc[tid] += sc[tid + o]; }
    __syncthreads();
  }
  if (tid == 0)
    c_top[b] = sf[0] + (float)(K - (int)sc[0]) * __uint_as_float(t);
}

// Kernel 6: deterministic final reduction -> out[0]=loss_l/N, out[1]=loss_c/N.
__global__ void __launch_bounds__(256) k_final(
    const float* __restrict__ partial_l, int nL,
    const float* __restrict__ partial_c, int nC,
    const float* __restrict__ c_top, const int* __restrict__ num_pos,
    int B, float* __restrict__ out) {
  __shared__ float sf[256];
  __shared__ int si[256];
  int tid = threadIdx.x;
  float sl = 0.f, scc = 0.f;
  int np = 0;
  for (int i = tid; i < nL; i += 256) sl += partial_l[i];
  for (int i = tid; i < nC; i += 256) scc += partial_c[i];
  for (int i = tid; i < B; i += 256) { scc += c_top[i]; np += num_pos[i]; }
  sf[tid] = sl; si[tid] = np;
  __syncthreads();
  for (int o = 128; o > 0; o >>= 1) {
    if (tid < o) { sf[tid] += sf[tid + o]; si[tid] += si[tid + o]; }
    __syncthreads();
  }
  float lossl = sf[0];
  int N = si[0];
  __syncthreads();
  sf[tid] = scc;
  __syncthreads();
  for (int o = 128; o > 0; o >>= 1) {
    if (tid < o) sf[tid] += sf[tid + o];
    __syncthreads();
  }
  if (tid == 0) {
    float Nf = (float)(N > 0 ? N : 1);
    out[0] = lossl / Nf;
    out[1] = sf[0] / Nf;
  }
}

extern "C" void kernel_launch(void* const* d_in, const int* in_sizes, int n_in,
                              void* d_out, int out_size, void* d_ws, size_t ws_size,
                              hipStream_t stream) {
  (void)n_in; (void)out_size; (void)ws_size;
  const float* loc_data  = (const float*)d_in[0];
  const float* conf_data = (const float*)d_in[1];
  const float* priors    = (const float*)d_in[2];
  const float* targets   = (const float*)d_in[3];

  int P = in_sizes[2] / 4;                                   // priors: (P,4)
  int B = (int)((long long)in_sizes[0] / (4LL * P));         // loc: (B,P,4)
  int C = (int)((long long)in_sizes[1] / ((long long)B * P)); // conf: (B,P,C)
  int NOBJ = in_sizes[3] / (B * 5);                          // targets: (B,NOBJ,5)
  long long R = (long long)B * P;

  char* ws = (char*)d_ws;
  size_t off = 0;
  auto carve = [&](size_t bytes) -> void* {
    void* ptr = (void*)(ws + off);
    off += (bytes + 255) & ~(size_t)255;
    return ptr;
  };
  float* bto = (float*)carve((size_t)R * sizeof(float));   // later reused as cem
  int* bti   = (int*)carve((size_t)R * sizeof(int));       // later reused as conf_t
  unsigned long long* bpk = (unsigned long long*)carve((size_t)B * NOBJ * 8);
  int* num_pos = (int*)carve((size_t)B * sizeof(int));
  int nblk = (P + 255) / 256;
  float* partial_l = (float*)carve((size_t)B * nblk * sizeof(float));
  const int NBLK4 = 4096;
  float* partial_c = (float*)carve((size_t)NBLK4 * sizeof(float));
  float* c_top = (float*)carve((size_t)B * sizeof(float));

  int nk = B * NOBJ;
  int initn = nk > B ? nk : B;
  k_init<<<(initn + 255) / 256, 256, 0, stream>>>(bpk, num_pos, nk, B);

  dim3 gbp(nblk, B);
  k_match<<<gbp, 256, 0, stream>>>(priors, targets, bto, bti, bpk, P, NOBJ);
  k_force<<<(B + 63) / 64, 64, 0, stream>>>(bpk, bto, bti, P, NOBJ, B);
  k_encode<<<gbp, 256, 0, stream>>>(priors, targets, loc_data, bto, bti,
                                    partial_l, num_pos, P, NOBJ);
  // bti now holds conf_t; bto is free -> reuse as ce_mine (cem).
  k_ce<<<NBLK4, 256, 0, stream>>>(conf_data, bti, bto, partial_c, R, C);
  k_select<<<B, 256, 0, stream>>>(bto, num_pos, c_top, P);
  k_final<<<1, 256, 0, stream>>>(partial_l, B * nblk, partial_c, NBLK4,
                                 c_top, num_pos, B, (float*)d_out);
}